// GraphSAGEModel_68676527063150
// MI455X (gfx1250) — compile-verified
//
#include <hip/hip_runtime.h>
#include <math.h>

#define N_NODES 100000
#define N_EDGES 1600000
#define DIM_IN  64
#define DIM_H   64
#define DIM_OUT 16

typedef __attribute__((ext_vector_type(2))) float v2f;
typedef __attribute__((ext_vector_type(8))) float v8f;

// ---------------------------------------------------------------------------
// Zero-fill scratch (grid-stride not needed; sized exactly)
// ---------------------------------------------------------------------------
__global__ __launch_bounds__(256) void zero_f32(float* __restrict__ p, int n) {
    int i = blockIdx.x * 256 + threadIdx.x;
    if (i < n) p[i] = 0.0f;
}

// ---------------------------------------------------------------------------
// Edge scatter: agg[dst] += feat[src]  (+ deg[dst] += 1 on first layer)
// 16 threads per edge, each moving a float4 (64 floats / edge).
// Feature table (25.6MB) and agg (25.6MB) are L2-resident on MI455X (192MB L2).
// ---------------------------------------------------------------------------
__global__ __launch_bounds__(256) void sage_scatter(
    const float* __restrict__ feat, const int* __restrict__ ei,
    float* __restrict__ agg, float* __restrict__ deg, int addDeg)
{
    int tid = blockIdx.x * 256 + threadIdx.x;
    if (tid >= N_EDGES * 16) return;
    int e    = tid >> 4;
    int part = tid & 15;
    int src = ei[e];
    int dst = ei[N_EDGES + e];

    const float4 v = *(const float4*)(feat + (size_t)src * 64 + part * 4);
    float* dp = agg + (size_t)dst * 64 + part * 4;
    atomicAdd(dp + 0, v.x);
    atomicAdd(dp + 1, v.y);
    atomicAdd(dp + 2, v.z);
    atomicAdd(dp + 3, v.w);
    if (addDeg && part == 0) atomicAdd(deg + dst, 1.0f);
}

// ---------------------------------------------------------------------------
// Layer-1 GEMM: h = relu(mean @ W1l + x @ W1r + b1)   [N,64]
// One 16-row tile per block; 4 waves, each owns one 16-col tile.
// v_wmma_f32_16x16x4_f32, K marched in steps of 4, fp32 end-to-end.
// A frag: lane<16 -> (K=k,k+1) of row lane; lane>=16 -> (K=k+2,k+3).
// B frag: lane<16 -> (K=k,k+1) of col lane; lane>=16 -> (K=k+2,k+3).
// ---------------------------------------------------------------------------
__global__ __launch_bounds__(128) void sage_gemm1(
    const float* __restrict__ x,   const float* __restrict__ agg,
    const float* __restrict__ deg, const float* __restrict__ W1l,
    const float* __restrict__ W1r, const float* __restrict__ b1,
    float* __restrict__ h)
{
    const int rowBase = blockIdx.x * 16;        // 6250 tiles cover 100000 rows
    const int wave    = threadIdx.x >> 5;       // col tile 0..3
    const int lane    = threadIdx.x & 31;
    const int colBase = wave * 16;

    const int lrow  = lane & 15;                // A row this lane feeds
    const int khalf = (lane >> 4) * 2;          // K sub-offset: 0 or 2
    const int ncol  = lane & 15;                // B/D column this lane feeds

    const float dinv = 1.0f / fmaxf(deg[rowBase + lrow], 1.0f);
    const float* arow_m = agg + (size_t)(rowBase + lrow) * 64;
    const float* arow_x = x   + (size_t)(rowBase + lrow) * 64;

    v8f c = {};
    // mean @ W1l
    for (int k = 0; k < 64; k += 4) {
        v2f a, b;
        a.x = arow_m[k + khalf + 0] * dinv;
        a.y = arow_m[k + khalf + 1] * dinv;
        const float* bp = W1l + (size_t)(k + khalf) * 64 + colBase + ncol;
        b.x = bp[0];
        b.y = bp[64];
        c = __builtin_amdgcn_wmma_f32_16x16x4_f32(false, a, false, b,
                                                  (short)0, c, false, false);
    }
    // x @ W1r
    for (int k = 0; k < 64; k += 4) {
        v2f a, b;
        a.x = arow_x[k + khalf + 0];
        a.y = arow_x[k + khalf + 1];
        const float* bp = W1r + (size_t)(k + khalf) * 64 + colBase + ncol;
        b.x = bp[0];
        b.y = bp[64];
        c = __builtin_amdgcn_wmma_f32_16x16x4_f32(false, a, false, b,
                                                  (short)0, c, false, false);
    }

    // bias + relu + store. D layout: VGPR r -> row r (lanes 0-15) / r+8 (16-31)
    const float bias = b1[colBase + ncol];
    const int rofs = (lane >> 4) * 8;
    #pragma unroll
    for (int r = 0; r < 8; ++r) {
        float v = fmaxf(c[r] + bias, 0.0f);
        h[(size_t)(rowBase + r + rofs) * 64 + colBase + ncol] = v;
    }
}

// ---------------------------------------------------------------------------
// Layer-2 GEMM + fused log_softmax: out = logsoftmax(mean2@W2l + h@W2r + b2)
// One wave per 16-row tile (DIM_OUT=16 -> single col tile). Pre-softmax tile
// is staged in LDS so lanes 0-15 can each reduce one row of 16.
// ---------------------------------------------------------------------------
__global__ __launch_bounds__(32) void sage_gemm2(
    const float* __restrict__ h,   const float* __restrict__ agg,
    const float* __restrict__ deg, const float* __restrict__ W2l,
    const float* __restrict__ W2r, const float* __restrict__ b2,
    float* __restrict__ out)
{
    __shared__ float tile[16][17];

    const int rowBase = blockIdx.x * 16;
    const int lane  = threadIdx.x;
    const int lrow  = lane & 15;
    const int khalf = (lane >> 4) * 2;
    const int ncol  = lane & 15;

    const float dinv = 1.0f / fmaxf(deg[rowBase + lrow], 1.0f);
    const float* arow_m = agg + (size_t)(rowBase + lrow) * 64;
    const float* arow_h = h   + (size_t)(rowBase + lrow) * 64;

    v8f c = {};
    // mean2 @ W2l   (W2l is [64,16] row-major)
    for (int k = 0; k < 64; k += 4) {
        v2f a, b;
        a.x = arow_m[k + khalf + 0] * dinv;
        a.y = arow_m[k + khalf + 1] * dinv;
        const float* bp = W2l + (size_t)(k + khalf) * 16 + ncol;
        b.x = bp[0];
        b.y = bp[16];
        c = __builtin_amdgcn_wmma_f32_16x16x4_f32(false, a, false, b,
                                                  (short)0, c, false, false);
    }
    // h @ W2r
    for (int k = 0; k < 64; k += 4) {
        v2f a, b;
        a.x = arow_h[k + khalf + 0];
        a.y = arow_h[k + khalf + 1];
        const float* bp = W2r + (size_t)(k + khalf) * 16 + ncol;
        b.x = bp[0];
        b.y = bp[16];
        c = __builtin_amdgcn_wmma_f32_16x16x4_f32(false, a, false, b,
                                                  (short)0, c, false, false);
    }

    const float bias = b2[ncol];
    const int rofs = (lane >> 4) * 8;
    #pragma unroll
    for (int r = 0; r < 8; ++r)
        tile[r + rofs][ncol] = c[r] + bias;

    __syncthreads();

    if (lane < 16) {
        const int row = lane;
        float m = tile[row][0];
        #pragma unroll
        for (int j = 1; j < 16; ++j) m = fmaxf(m, tile[row][j]);
        float s = 0.0f;
        #pragma unroll
        for (int j = 0; j < 16; ++j) s += __expf(tile[row][j] - m);
        const float lse = m + __logf(s);
        float* orow = out + (size_t)(rowBase + row) * 16;
        #pragma unroll
        for (int j = 0; j < 16; ++j) orow[j] = tile[row][j] - lse;
    }
}

// ---------------------------------------------------------------------------
extern "C" void kernel_launch(void* const* d_in, const int* in_sizes, int n_in,
                              void* d_out, int out_size, void* d_ws, size_t ws_size,
                              hipStream_t stream) {
    const float* x   = (const float*)d_in[0];
    const int*   ei  = (const int*)  d_in[1];
    const float* W1l = (const float*)d_in[2];
    const float* W1r = (const float*)d_in[3];
    const float* b1  = (const float*)d_in[4];
    const float* W2l = (const float*)d_in[5];
    const float* W2r = (const float*)d_in[6];
    const float* b2  = (const float*)d_in[7];
    float* out = (float*)d_out;

    // workspace layout: deg[N] | agg[N*64] | h[N*64]   (~51.7 MB)
    float* deg = (float*)d_ws;
    float* agg = deg + N_NODES;
    float* hbuf = agg + (size_t)N_NODES * 64;

    const int nZero1 = N_NODES + N_NODES * 64;   // deg + agg contiguous
    const int nZero2 = N_NODES * 64;
    const int scatterThreads = N_EDGES * 16;

    // ---- layer 1 ----
    zero_f32<<<(nZero1 + 255) / 256, 256, 0, stream>>>(deg, nZero1);
    sage_scatter<<<(scatterThreads + 255) / 256, 256, 0, stream>>>(
        x, ei, agg, deg, 1);
    sage_gemm1<<<N_NODES / 16, 128, 0, stream>>>(x, agg, deg, W1l, W1r, b1, hbuf);

    // ---- layer 2 ----
    zero_f32<<<(nZero2 + 255) / 256, 256, 0, stream>>>(agg, nZero2);
    sage_scatter<<<(scatterThreads + 255) / 256, 256, 0, stream>>>(
        hbuf, ei, agg, deg, 0);
    sage_gemm2<<<N_NODES / 16, 32, 0, stream>>>(hbuf, agg, deg, W2l, W2r, b2, out);
}